// MultiHeadAttentionLayer_45208825758053
// MI455X (gfx1250) — compile-verified
//
#include <hip/hip_runtime.h>
#include <hip/hip_bf16.h>

// ---------------------------------------------------------------------------
// Problem constants (match reference)
// ---------------------------------------------------------------------------
#define NNODES   20000
#define E_FULL   640000
#define E_G      160000
#define IN_DIM   128
#define OUT_DIM  16
#define NHEAD    8
#define HD       128          // OUT_DIM * NHEAD
#define EDGE_DIM 128
#define INV_SCALE 0.25f       // 1/sqrt(16)

typedef __attribute__((ext_vector_type(2))) float v2f;
typedef __attribute__((ext_vector_type(8))) float v8f;

// ---------------------------------------------------------------------------
// fp32 WMMA GEMM:  C[M x Nc] = (A [+ A2]) [M x K] @ B [K x Nc] + bias
// One wave32 computes one 16x16 tile of C via V_WMMA_F32_16X16X4_F32.
//  - K, LDA, LDB, LDC are compile-time: all addressing folds into the
//    global_load 24-bit immediate offsets (no v_add_nc_u64 chains, no
//    per-load s_wait_xcnt serialization).
//  - M must be a multiple of 16 (true for 20000 / 160000).
//  - CHECK_N handles the one Nc==8 GEMM with branch-free selects (no exec-mask
//    churn; EXEC stays all-1s around every WMMA as the ISA requires).
//  - Two accumulators break the WMMA->WMMA RAW chain (even/odd K halves).
// ---------------------------------------------------------------------------
template<int K, int LDA, int LDB, int LDC, bool HAS_A2, bool CHECK_N>
__global__ void gemm_wmma_f32_kernel(const float* __restrict__ A,
                                     const float* __restrict__ A2,
                                     const float* __restrict__ B,
                                     const float* __restrict__ bias,
                                     float* __restrict__ C,
                                     int M, int Nc)
{
    const int gtid = blockIdx.x * blockDim.x + threadIdx.x;
    const int wave = gtid >> 5;          // wave32
    const int lane = gtid & 31;

    const int ntn = (Nc + 15) >> 4;
    if (wave >= (M >> 4) * ntn) return;  // wave-uniform -> EXEC stays full

    const int tm = wave / ntn;
    const int tn = wave % ntn;
    const int m0 = tm << 4;
    const int n0 = tn << 4;

    const int half = lane >> 4;          // selects the K-pair of the fragment
    const int l16  = lane & 15;
    const int am   = m0 + l16;           // A fragment row   (M index)
    const int bn   = n0 + l16;           // B fragment col / C col (N index)

    bool bok  = true;
    int  bcol = bn;
    if (CHECK_N) {                       // branch-free select, no divergence
        bok  = (bn < Nc);
        bcol = bok ? bn : 0;
    }

    const float* __restrict__ Arow  = A + (size_t)am * LDA + half * 2;
    const float* __restrict__ A2row = HAS_A2 ? (A2 + (size_t)am * LDA + half * 2)
                                             : nullptr;
    const float* __restrict__ Bcol  = B + half * 2 * LDB + bcol;

    v8f acc0 = {};
    v8f acc1 = {};
#pragma unroll
    for (int k = 0; k < K; k += 8) {
        v2f a0 = *(const v2f*)(Arow + k);        // imm offset k*4
        v2f a1 = *(const v2f*)(Arow + k + 4);
        if (HAS_A2) {
            v2f t0 = *(const v2f*)(A2row + k);
            v2f t1 = *(const v2f*)(A2row + k + 4);
            a0.x += t0.x; a0.y += t0.y;
            a1.x += t1.x; a1.y += t1.y;
        }
        v2f b0, b1;
        b0.x = Bcol[(k + 0) * LDB];              // imm offset (k+0)*LDB*4
        b0.y = Bcol[(k + 1) * LDB];
        b1.x = Bcol[(k + 4) * LDB];
        b1.y = Bcol[(k + 5) * LDB];
        if (CHECK_N && !bok) {
            b0.x = 0.f; b0.y = 0.f; b1.x = 0.f; b1.y = 0.f;
        }
        // 8 args: (neg_a, A, neg_b, B, c_mod, C, reuse_a, reuse_b)
        acc0 = __builtin_amdgcn_wmma_f32_16x16x4_f32(false, a0, false, b0,
                                                     (short)0, acc0, false, false);
        acc1 = __builtin_amdgcn_wmma_f32_16x16x4_f32(false, a1, false, b1,
                                                     (short)0, acc1, false, false);
    }

    if (!CHECK_N || bok) {
        const float bv = bias[bcol];
        // C/D layout: VGPR v -> M = v + 8*(lane/16); single base + imm offsets
        float* __restrict__ Crow = C + (size_t)(m0 + 8 * half) * LDC + bn;
#pragma unroll
        for (int v = 0; v < 8; ++v) {
            Crow[v * LDC] = acc0[v] + acc1[v] + bv;
        }
    }
}

// ---------------------------------------------------------------------------
// Edge score: score[e,h] = exp(clip(<Kh[src],Qh[dst]>_h / 4, -5, 5) * adj2[e])
//                          + rel[e]
// ---------------------------------------------------------------------------
__global__ void edge_score_kernel(const float* __restrict__ Qh,
                                  const float* __restrict__ Kh,
                                  const int* __restrict__ src,
                                  const int* __restrict__ dst,
                                  const float* __restrict__ adj2,
                                  const float* __restrict__ rel,
                                  float* __restrict__ score)
{
    const int t = blockIdx.x * blockDim.x + threadIdx.x;
    if (t >= E_FULL * NHEAD) return;
    const int e = t >> 3;
    const int h = t & 7;
    const int s = src[e];
    const int d = dst[e];
    const float4* kp = (const float4*)(Kh + (size_t)s * HD + h * OUT_DIM);
    const float4* qp = (const float4*)(Qh + (size_t)d * HD + h * OUT_DIM);
    float dot = 0.0f;
#pragma unroll
    for (int i = 0; i < 4; ++i) {
        const float4 kv = kp[i];
        const float4 qv = qp[i];
        dot += kv.x * qv.x + kv.y * qv.y + kv.z * qv.z + kv.w * qv.w;
    }
    dot *= INV_SCALE;
    const float c = fminf(fmaxf(dot, -5.0f), 5.0f);
    score[t] = __expf(c * adj2[e]) + rel[e];
}

// gs[i,h] = score[edge_map[i], h]
__global__ void gather_gs_kernel(const float* __restrict__ score,
                                 const int* __restrict__ edge_map,
                                 float* __restrict__ gs)
{
    const int t = blockIdx.x * blockDim.x + threadIdx.x;
    if (t >= E_G * NHEAD) return;
    const int i = t >> 3;
    const int h = t & 7;
    gs[t] = score[(size_t)edge_map[i] * NHEAD + h];
}

// score[edge_map[i], h] = clip(gs[i,h], -5, 5) + proj_e[i,h]
__global__ void g_update_kernel(const float* __restrict__ gs,
                                const float* __restrict__ proj_e,
                                const int* __restrict__ edge_map,
                                float* __restrict__ score)
{
    const int t = blockIdx.x * blockDim.x + threadIdx.x;
    if (t >= E_G * NHEAD) return;
    const int i = t >> 3;
    const int h = t & 7;
    const float s = gs[t];
    score[(size_t)edge_map[i] * NHEAD + h] =
        fminf(fmaxf(s, -5.0f), 5.0f) + proj_e[t];
}

// score = exp(clip(score, -5, 5))
__global__ void exp_kernel(float* __restrict__ score, int n)
{
    const int t = blockIdx.x * blockDim.x + threadIdx.x;
    if (t >= n) return;
    const float s = score[t];
    score[t] = __expf(fminf(fmaxf(s, -5.0f), 5.0f));
}

__global__ void zero_kernel(float* __restrict__ p, int n)
{
    const int t = blockIdx.x * blockDim.x + threadIdx.x;
    if (t < n) p[t] = 0.0f;
}

// wV[dst, c] += Vh[src, c] * score[e, c/16] ; z[dst, h] += score[e, h]
__global__ void aggregate_kernel(const float* __restrict__ Vh,
                                 const float* __restrict__ score,
                                 const int* __restrict__ src,
                                 const int* __restrict__ dst,
                                 float* __restrict__ wV,
                                 float* __restrict__ z)
{
    const long long t = (long long)blockIdx.x * blockDim.x + threadIdx.x;
    if (t >= (long long)E_FULL * HD) return;
    const int e = (int)(t >> 7);
    const int c = (int)(t & 127);
    const int h = c >> 4;
    const float s = score[(size_t)e * NHEAD + h];
    const float v = Vh[(size_t)src[e] * HD + c];
    atomicAdd(&wV[(size_t)dst[e] * HD + c], v * s);
    if ((c & 15) == 0) atomicAdd(&z[(size_t)dst[e] * NHEAD + h], s);
}

// hn = wV / (z + 1e-6)
__global__ void norm_kernel(const float* __restrict__ wV,
                            const float* __restrict__ z,
                            float* __restrict__ hn)
{
    const int t = blockIdx.x * blockDim.x + threadIdx.x;
    if (t >= NNODES * HD) return;
    const int n = t >> 7;
    const int c = t & 127;
    const int h = c >> 4;
    hn[t] = wV[t] / (z[(size_t)n * NHEAD + h] + 1e-6f);
}

// ---------------------------------------------------------------------------
// Host side
// ---------------------------------------------------------------------------
static inline size_t align256(size_t x) { return (x + 255) & ~(size_t)255; }

static inline int gemm_grid(int M, int Nc)
{
    const long long waves = (long long)(M / 16) * ((Nc + 15) / 16);
    return (int)((waves * 32 + 255) / 256);
}

extern "C" void kernel_launch(void* const* d_in, const int* in_sizes, int n_in,
                              void* d_out, int out_size, void* d_ws, size_t ws_size,
                              hipStream_t stream)
{
    // Inputs in setup_inputs() order
    const float* h        = (const float*)d_in[0];
    const float* e        = (const float*)d_in[1];
    const float* adj2     = (const float*)d_in[2];
    const float* rel      = (const float*)d_in[3];
    const int*   src      = (const int*)d_in[4];
    const int*   dst      = (const int*)d_in[5];
    const int*   edge_map = (const int*)d_in[6];
    const float* Wq  = (const float*)d_in[7];   const float* bq  = (const float*)d_in[8];
    const float* Wk  = (const float*)d_in[9];   const float* bk  = (const float*)d_in[10];
    const float* Wv  = (const float*)d_in[11];  const float* bv  = (const float*)d_in[12];
    const float* Wpe = (const float*)d_in[13];  const float* bpe = (const float*)d_in[14];
    const float* Wap = (const float*)d_in[15];  const float* bap = (const float*)d_in[16];
    const float* Wout  = (const float*)d_in[17]; const float* bout  = (const float*)d_in[18];
    const float* Woute = (const float*)d_in[19]; const float* boute = (const float*)d_in[20];

    float* out = (float*)d_out;
    float* h_out = out;                         // [NNODES x HD]
    float* e_out = out + (size_t)NNODES * HD;   // [E_G x EDGE_DIM]

    // Workspace carving
    char* ws = (char*)d_ws;
    size_t off = 0;
    float* Qh     = (float*)(ws + off); off = align256(off + (size_t)NNODES * HD * 4);
    float* Kh     = (float*)(ws + off); off = align256(off + (size_t)NNODES * HD * 4);
    float* Vh     = (float*)(ws + off); off = align256(off + (size_t)NNODES * HD * 4);
    float* hn     = (float*)(ws + off); off = align256(off + (size_t)NNODES * HD * 4);
    float* proj_e = (float*)(ws + off); off = align256(off + (size_t)E_G * NHEAD * 4);
    float* score  = (float*)(ws + off); off = align256(off + (size_t)E_FULL * NHEAD * 4);
    float* gs     = (float*)(ws + off); off = align256(off + (size_t)E_G * NHEAD * 4);
    float* e_tmp  = (float*)(ws + off); off = align256(off + (size_t)E_G * EDGE_DIM * 4);
    // Aliased regions: Qh/Kh dead after edge_score_kernel
    float* wV = Qh;                       // [NNODES x HD]
    float* z  = Kh;                       // [NNODES x NHEAD]

    const int B = 256;

    // 1) Q/K/V projections (WMMA GEMMs, K=128)
    gemm_wmma_f32_kernel<128, 128, 128, 128, false, false>
        <<<gemm_grid(NNODES, HD), B, 0, stream>>>(h, nullptr, Wq, bq, Qh, NNODES, HD);
    gemm_wmma_f32_kernel<128, 128, 128, 128, false, false>
        <<<gemm_grid(NNODES, HD), B, 0, stream>>>(h, nullptr, Wk, bk, Kh, NNODES, HD);
    gemm_wmma_f32_kernel<128, 128, 128, 128, false, false>
        <<<gemm_grid(NNODES, HD), B, 0, stream>>>(h, nullptr, Wv, bv, Vh, NNODES, HD);

    // 2) proj_e = e @ Wpe + bpe   [E_G x 8]  (N=8 -> CHECK_N specialization)
    gemm_wmma_f32_kernel<128, 128, 8, 8, false, true>
        <<<gemm_grid(E_G, NHEAD), B, 0, stream>>>(e, nullptr, Wpe, bpe, proj_e, E_G, NHEAD);

    // 3) edge scores on full graph
    {
        const int n = E_FULL * NHEAD;
        edge_score_kernel<<<(n + B - 1) / B, B, 0, stream>>>(
            Qh, Kh, src, dst, adj2, rel, score);
    }

    // 4) gather g-scores, attn_proj GEMM (K=8), in-place score update, exp
    {
        const int n = E_G * NHEAD;
        gather_gs_kernel<<<(n + B - 1) / B, B, 0, stream>>>(score, edge_map, gs);
    }
    gemm_wmma_f32_kernel<8, 8, 128, 128, false, false>
        <<<gemm_grid(E_G, EDGE_DIM), B, 0, stream>>>(gs, nullptr, Wap, bap, e_tmp,
                                                     E_G, EDGE_DIM);
    {
        const int n = E_G * NHEAD;
        g_update_kernel<<<(n + B - 1) / B, B, 0, stream>>>(gs, proj_e, edge_map, score);
    }
    {
        const int n = E_FULL * NHEAD;
        exp_kernel<<<(n + B - 1) / B, B, 0, stream>>>(score, n);
    }

    // 5) weighted aggregation (atomics into wV, z)
    {
        const int n1 = NNODES * HD;
        zero_kernel<<<(n1 + B - 1) / B, B, 0, stream>>>(wV, n1);
        const int n2 = NNODES * NHEAD;
        zero_kernel<<<(n2 + B - 1) / B, B, 0, stream>>>(z, n2);
        const long long n3 = (long long)E_FULL * HD;
        aggregate_kernel<<<(int)((n3 + B - 1) / B), B, 0, stream>>>(
            Vh, score, src, dst, wV, z);
    }

    // 6) normalize and output projections (WMMA GEMMs straight into d_out)
    {
        const int n = NNODES * HD;
        norm_kernel<<<(n + B - 1) / B, B, 0, stream>>>(wV, z, hn);
    }
    gemm_wmma_f32_kernel<128, 128, 128, 128, false, false>
        <<<gemm_grid(NNODES, HD), B, 0, stream>>>(hn, nullptr, Wout, bout, h_out,
                                                  NNODES, HD);
    // e_out = (e_tmp + e) @ Woute + boute   (A2 addend fused into A-load)
    gemm_wmma_f32_kernel<128, 128, 128, 128, true, false>
        <<<gemm_grid(E_G, EDGE_DIM), B, 0, stream>>>(e_tmp, e, Woute, boute, e_out,
                                                     E_G, EDGE_DIM);
}